// MMSEPICDetector_55018531062642
// MI455X (gfx1250) — compile-verified
//
#include <hip/hip_runtime.h>
#include <math.h>

typedef __attribute__((ext_vector_type(2))) float v2f;
typedef __attribute__((ext_vector_type(8))) float v8f;

#define WAVES_PER_BLOCK 8
#define EPSV 1e-4f
#define INV_SQRT10 0.3162277660168379f

// Gray-mapped QAM16 (MSB-first bit labels), un-normalized PAM amplitudes.
// re depends on bits (b0,b2), im on (b1,b3): (1-2a)*(1+2b)
__device__ const float kPre[16] = {1,1,3,3,1,1,3,3,-1,-1,-3,-3,-1,-1,-3,-3};
__device__ const float kPim[16] = {1,3,1,3,-1,-3,-1,-3,1,3,1,3,-1,-3,-1,-3};

struct WS {
  union {
    struct ChT { float Are[16][17]; float Aim[16][17]; } ch;  // S -> Cholesky L
    float aug[16][33];                                        // [a | I] Gauss-Jordan
  } u;
  float Wre[16][10], Wim[16][10];   // whitened [h(0..7) | y(8)]
  float gre[8][9],  gim[8][9];      // complex Gram g = h^H h
  float gr[16][17];                 // real Gram [[Re,-Im],[Im,Re]]
  float yptr[16][17];               // ypt_r (rows duplicated)
  float ymfre[8], ymfim[8];
  float xre[8], xim[8], varL[8];
  float gxre[8], gxim[8];
  float muL[16], znumL[16];
  float rhoL[8];
  float llr_aL[8][4], llr_dL[8][4];
};

__device__ __forceinline__ float log_sigmoid(float x) {
  // log(1/(1+e^-x)) = min(x,0) - log1p(e^{-|x|})
  return fminf(x, 0.0f) - log1pf(__expf(-fabsf(x)));
}

__global__ __launch_bounds__(256) void mmse_pic_kernel(
    const float* __restrict__ y_re, const float* __restrict__ y_im,
    const float* __restrict__ h_re, const float* __restrict__ h_im,
    const float* __restrict__ prior,
    const float* __restrict__ s_re, const float* __restrict__ s_im,
    float* __restrict__ out)
{
  __shared__ WS ws[WAVES_PER_BLOCK];
  const int lane = threadIdx.x & 31;
  const int wid  = threadIdx.x >> 5;
  const int b    = blockIdx.x * WAVES_PER_BLOCK + wid;
  WS& w = ws[wid];

  // ---------------- load S (16x16 complex), coalesced ----------------
  #pragma unroll
  for (int t = 0; t < 8; ++t) {
    int idx = lane + 32*t;
    int i = idx >> 4, j = idx & 15;
    w.u.ch.Are[i][j] = s_re[b*256 + idx];
    w.u.ch.Aim[i][j] = s_im[b*256 + idx];
  }
  // ---------------- load H (16x8 complex), y -> col 8 ----------------
  #pragma unroll
  for (int t = 0; t < 4; ++t) {
    int idx = lane + 32*t;
    int m = idx >> 3, c = idx & 7;
    w.Wre[m][c] = h_re[b*128 + idx];
    w.Wim[m][c] = h_im[b*128 + idx];
  }
  if (lane < 16) {
    w.Wre[lane][8] = y_re[b*16 + lane];
    w.Wim[lane][8] = y_im[b*16 + lane];
  }
  const int s_out = lane >> 2, k_out = lane & 3;   // lane = s*4 + k
  float my_lld = prior[b*32 + lane];
  w.llr_dL[s_out][k_out] = my_lld;
  __syncthreads();

  // ---------------- complex Cholesky (lower, in place), lane = row ----
  const int r = lane;
  for (int k = 0; k < 16; ++k) {
    float dk  = w.u.ch.Are[k][k];
    float sdk = sqrtf(dk);
    float isd = 1.0f / sdk;
    if (r == k) { w.u.ch.Are[k][k] = sdk; w.u.ch.Aim[k][k] = 0.0f; }
    if (r < 16 && r > k) { w.u.ch.Are[r][k] *= isd; w.u.ch.Aim[r][k] *= isd; }
    __syncthreads();
    if (r < 16 && r > k) {
      float lre = w.u.ch.Are[r][k], lim = w.u.ch.Aim[r][k];
      for (int j = k+1; j <= r; ++j) {
        float cre = w.u.ch.Are[j][k], cim = w.u.ch.Aim[j][k];
        // A[r][j] -= L[r][k] * conj(L[j][k])
        w.u.ch.Are[r][j] -= lre*cre + lim*cim;
        w.u.ch.Aim[r][j] -= lim*cre - lre*cim;
      }
    }
    __syncthreads();
  }

  // ---------------- whiten: forward-solve L x = [h|y], lane = column --
  if (lane < 9) {
    const int c = lane;
    for (int m = 0; m < 16; ++m) {
      float accre = w.Wre[m][c], accim = w.Wim[m][c];
      for (int j = 0; j < m; ++j) {
        float lre = w.u.ch.Are[m][j], lim = w.u.ch.Aim[m][j];
        float wr = w.Wre[j][c], wi = w.Wim[j][c];
        accre -= lre*wr - lim*wi;
        accim -= lre*wi + lim*wr;
      }
      float inv = 1.0f / w.u.ch.Are[m][m];   // diag is real > 0
      w.Wre[m][c] = accre*inv;
      w.Wim[m][c] = accim*inv;
    }
  }
  __syncthreads();

  // ---------------- y_mf = h^H y ----------------
  if (lane < 8) {
    int s_ = lane;
    float ar = 0.0f, ai = 0.0f;
    #pragma unroll
    for (int m = 0; m < 16; ++m) {
      float hr = w.Wre[m][s_], hi = w.Wim[m][s_];
      float yr = w.Wre[m][8],  yi = w.Wim[m][8];
      ar += hr*yr + hi*yi;
      ai += hr*yi - hi*yr;
    }
    w.ymfre[s_] = ar; w.ymfim[s_] = ai;
  }
  // ---------------- g = h^H h (8x8 complex), 2 entries per lane -------
  #pragma unroll
  for (int t = 0; t < 2; ++t) {
    int idx = lane + 32*t;
    int i = idx >> 3, j = idx & 7;
    float ar = 0.0f, ai = 0.0f;
    #pragma unroll
    for (int m = 0; m < 16; ++m) {
      float xr = w.Wre[m][i], xi = w.Wim[m][i];
      float yr = w.Wre[m][j], yi = w.Wim[m][j];
      ar += xr*yr + xi*yi;
      ai += xr*yi - xi*yr;
    }
    w.gre[i][j] = ar; w.gim[i][j] = ai;
  }
  __syncthreads();
  // ---------------- gr = [[Re g, -Im g],[Im g, Re g]] -----------------
  #pragma unroll
  for (int t = 0; t < 8; ++t) {
    int idx = lane + 32*t;
    int i = idx >> 4, j = idx & 15;
    float v;
    if (i < 8) v = (j < 8) ? w.gre[i][j]   : -w.gim[i][j-8];
    else       v = (j < 8) ? w.gim[i-8][j] :  w.gre[i-8][j-8];
    w.gr[i][j] = v;
  }
  __syncthreads();

  // ================== MMSE-PIC iterations ==================
  float my_lla = 0.0f;
  for (int iter = 0; iter < 2; ++iter) {
    my_lla = my_lld;
    w.llr_aL[s_out][k_out] = my_lla;
    __syncthreads();

    float lg[16];   // per-point symbol log-priors (lanes 0..7)
    // ---- moments: soft symbols from llr_a ----
    if (lane < 8) {
      int s_ = lane;
      float l0 = w.llr_aL[s_][0], l1 = w.llr_aL[s_][1];
      float l2 = w.llr_aL[s_][2], l3 = w.llr_aL[s_][3];
      #pragma unroll
      for (int p = 0; p < 16; ++p) {
        lg[p] = log_sigmoid((p & 8) ? l0 : -l0)
              + log_sigmoid((p & 4) ? l1 : -l1)
              + log_sigmoid((p & 2) ? l2 : -l2)
              + log_sigmoid((p & 1) ? l3 : -l3);
      }
      float mx = lg[0];
      #pragma unroll
      for (int p = 1; p < 16; ++p) mx = fmaxf(mx, lg[p]);
      float sum = 0.0f, mre = 0.0f, mim = 0.0f, e2 = 0.0f;
      #pragma unroll
      for (int p = 0; p < 16; ++p) {
        float e = __expf(lg[p] - mx);
        float pr = kPre[p]*INV_SQRT10, pi = kPim[p]*INV_SQRT10;
        sum += e; mre += e*pr; mim += e*pi; e2 += e*(pr*pr + pi*pi);
      }
      float invs = 1.0f / sum;
      mre *= invs; mim *= invs; e2 *= invs;
      w.xre[s_] = mre; w.xim[s_] = mim;
      w.varL[s_] = e2 - (mre*mre + mim*mim);
    }
    __syncthreads();

    // ---- gx = g @ x_hat ----
    if (lane < 8) {
      int i = lane;
      float ar = 0.0f, ai = 0.0f;
      #pragma unroll
      for (int j = 0; j < 8; ++j) {
        float gre_ = w.gre[i][j], gim_ = w.gim[i][j];
        float xr = w.xre[j], xi = w.xim[j];
        ar += gre_*xr - gim_*xi;
        ai += gre_*xi + gim_*xr;
      }
      w.gxre[i] = ar; w.gxim[i] = ai;
    }
    __syncthreads();

    // ---- ypt_r: ypt[i][j] = y_mf[j] + g[j][i]*x[i] - (g x)[j] --------
    #pragma unroll
    for (int t = 0; t < 2; ++t) {
      int idx = lane + 32*t;
      int i = idx >> 3, j = idx & 7;        // stream i, obs j
      float gre_ = w.gre[j][i], gim_ = w.gim[j][i];
      float xr = w.xre[i], xi = w.xim[i];
      float tre = w.ymfre[j] + (gre_*xr - gim_*xi) - w.gxre[j];
      float tim = w.ymfim[j] + (gre_*xi + gim_*xr) - w.gxim[j];
      w.yptr[i][j]     = tre; w.yptr[i][j+8]   = tim;
      w.yptr[i+8][j]   = tre; w.yptr[i+8][j+8] = tim;
    }
    // ---- build augmented [a | I], a = gr*diag(var2) + I --------------
    {
      int rr = lane & 15, hf = lane >> 4, base = hf*16;
      #pragma unroll
      for (int c = 0; c < 16; ++c) {
        int col = base + c;
        float v;
        if (col < 16) v = w.gr[rr][col] * w.varL[col & 7] + ((rr == col) ? 1.0f : 0.0f);
        else          v = ((col - 16) == rr) ? 1.0f : 0.0f;
        w.u.aug[rr][col] = v;
      }
    }
    __syncthreads();

    // ---- Gauss-Jordan inverse (no pivoting; a is diag-dominant) ------
    {
      int rr = lane & 15, hf = lane >> 4, base = hf*16;
      for (int k = 0; k < 16; ++k) {
        float pinv = 1.0f / w.u.aug[k][k];
        __syncthreads();
        if (rr == k) {
          #pragma unroll
          for (int c = 0; c < 16; ++c) w.u.aug[k][base+c] *= pinv;
        }
        __syncthreads();
        float f = w.u.aug[rr][k];
        if (rr != k) {
          #pragma unroll
          for (int c = 0; c < 16; ++c)
            w.u.aug[rr][base+c] -= f * w.u.aug[k][base+c];
        }
        __syncthreads();
      }
    }

    // ---- WMMA: P1 = Ainv @ gr, P2 = Ainv @ yptr^T; take diagonals ----
    // f32 16x16x4: A lanes 0-15 hold K=0,1; lanes 16-31 hold K=2,3.
    {
      int nn = lane & 15, hf = lane >> 4;
      v8f p1 = {}; v8f p2 = {};
      #pragma unroll
      for (int kb = 0; kb < 4; ++kb) {
        int k0 = kb*4 + hf*2;
        v2f af, b1, b2;
        af.x = w.u.aug[nn][16 + k0];       // Ainv[m=nn][k0]
        af.y = w.u.aug[nn][16 + k0 + 1];
        b1.x = w.gr[k0][nn];   b1.y = w.gr[k0+1][nn];     // B[k][n] = gr[k][n]
        b2.x = w.yptr[nn][k0]; b2.y = w.yptr[nn][k0+1];   // B[k][n] = yptr[n][k]
        p1 = __builtin_amdgcn_wmma_f32_16x16x4_f32(false, af, false, b1,
                                                   (short)0, p1, false, false);
        p2 = __builtin_amdgcn_wmma_f32_16x16x4_f32(false, af, false, b2,
                                                   (short)0, p2, false, false);
      }
      // D layout: vgpr v holds row (v + 8*hf), col = nn
      #pragma unroll
      for (int v = 0; v < 8; ++v) {
        int row = v + 8*hf;
        if (row == nn) { w.muL[nn] = p1[v]; w.znumL[nn] = p2[v]; }
      }
    }
    __syncthreads();

    // ---- unbiased estimates + post-SINR ----
    if (lane < 8) {
      int s_ = lane;
      float m1 = w.muL[s_], m2 = w.muL[s_ + 8];
      w.xre[s_] = w.znumL[s_]     / m1;
      w.xim[s_] = w.znumL[s_ + 8] / m2;
      w.rhoL[s_] = m1 / fmaxf(1.0f - w.varL[s_] * m1, EPSV);
    }
    __syncthreads();

    // ---- max-log demap with prior (reuses lg[] from this iteration) --
    if (lane < 8) {
      int s_ = lane;
      float zr = w.xre[s_], zi = w.xim[s_];
      float rho = w.rhoL[s_];                // 1/no_eff
      float b1m[4] = {-1e30f,-1e30f,-1e30f,-1e30f};
      float b0m[4] = {-1e30f,-1e30f,-1e30f,-1e30f};
      #pragma unroll
      for (int p = 0; p < 16; ++p) {
        float pr = kPre[p]*INV_SQRT10, pi = kPim[p]*INV_SQRT10;
        float dre = zr - pr, dim = zi - pi;
        float e = -(dre*dre + dim*dim) * rho + lg[p];
        #pragma unroll
        for (int k = 0; k < 4; ++k) {
          int bit = (p >> (3-k)) & 1;
          if (bit) b1m[k] = fmaxf(b1m[k], e);
          else     b0m[k] = fmaxf(b0m[k], e);
        }
      }
      #pragma unroll
      for (int k = 0; k < 4; ++k) w.llr_dL[s_][k] = b1m[k] - b0m[k];
    }
    __syncthreads();
    my_lld = w.llr_dL[s_out][k_out];
    __syncthreads();
  }

  // extrinsic LLRs, coalesced store
  out[b*32 + lane] = my_lld - my_lla;
}

extern "C" void kernel_launch(void* const* d_in, const int* in_sizes, int n_in,
                              void* d_out, int out_size, void* d_ws, size_t ws_size,
                              hipStream_t stream) {
  (void)n_in; (void)out_size; (void)d_ws; (void)ws_size;
  const float* y_re  = (const float*)d_in[0];
  const float* y_im  = (const float*)d_in[1];
  const float* h_re  = (const float*)d_in[2];
  const float* h_im  = (const float*)d_in[3];
  const float* prior = (const float*)d_in[4];
  const float* s_re  = (const float*)d_in[5];
  const float* s_im  = (const float*)d_in[6];
  float* out = (float*)d_out;

  int B = in_sizes[0] / 16;               // y_re is [B, 16]
  int blocks = B / WAVES_PER_BLOCK;       // one wave32 per batch item
  mmse_pic_kernel<<<blocks, 256, 0, stream>>>(y_re, y_im, h_re, h_im,
                                              prior, s_re, s_im, out);
}